// Net_69655779606898
// MI455X (gfx1250) — compile-verified
//
#include <hip/hip_runtime.h>
#include <hip/hip_bf16.h>

typedef __attribute__((ext_vector_type(2))) float v2f;
typedef __attribute__((ext_vector_type(8))) float v8f;

#define N_NODES 100000
#define N_EDGES 1600000
#define NTILES  (N_NODES / 16)   // 6250, exact

// ---------------- degree / normalization ----------------

__global__ __launch_bounds__(256) void k_deg_init(float* __restrict__ deg) {
    int i = blockIdx.x * 256 + threadIdx.x;
    if (i < N_NODES) deg[i] = 1.0f;   // self-loop weight
}

__global__ __launch_bounds__(256) void k_deg_accum(const int* __restrict__ dst,
                                                   const float* __restrict__ w,
                                                   float* __restrict__ deg) {
    int e = blockIdx.x * 256 + threadIdx.x;
    if (e < N_EDGES) atomicAdd(&deg[dst[e]], w[e]);
}

__global__ __launch_bounds__(256) void k_dis(float* __restrict__ d) {
    int i = blockIdx.x * 256 + threadIdx.x;
    if (i < N_NODES) d[i] = rsqrtf(d[i]);   // deg >= 1 always
}

__global__ __launch_bounds__(256) void k_norm(const int* __restrict__ src,
                                              const int* __restrict__ dst,
                                              const float* __restrict__ w,
                                              const float* __restrict__ dis,
                                              float* __restrict__ norm) {
    int e = blockIdx.x * 256 + threadIdx.x;
    if (e < N_EDGES) norm[e] = dis[src[e]] * w[e] * dis[dst[e]];
}

// ---------------- WMMA GEMMs (fp32 16x16x4) ----------------
// A 16x4 layout: lanes 0-15 rows 0-15 hold K=0,1 in v0,v1; lanes 16-31 hold K=2,3.
// B 4x16 layout (mirrored): lanes 0-15 cols, v0=K0 v1=K1; lanes 16-31 v0=K2 v1=K3.
// C/D: vgpr r, lanes 0-15 -> (M=r, N=lane), lanes 16-31 -> (M=r+8, N=lane-16).

// h[100000,64] = x[100000,128] @ W[128,64]
__global__ __launch_bounds__(128) void k_gemm1(const float* __restrict__ x,
                                               const float* __restrict__ W,
                                               float* __restrict__ h) {
    const int lane = threadIdx.x & 31;
    const int wave = threadIdx.x >> 5;
    const int tile = blockIdx.x * 4 + wave;
    if (tile >= NTILES) return;                 // wave-uniform: EXEC stays all-ones
    const int m  = lane & 15;
    const int hi = lane >> 4;
    const int row0 = tile * 16;

    v8f acc0 = {}, acc1 = {}, acc2 = {}, acc3 = {};
    const float* __restrict__ xr = x + (size_t)(row0 + m) * 128;

    for (int k = 0; k < 128; k += 4) {
        const int ka = k + 2 * hi;
        v2f a; a.x = xr[ka]; a.y = xr[ka + 1];
        const float* Wk0 = W + (size_t)ka * 64;
        v2f b0; b0.x = Wk0[m];      b0.y = Wk0[64 + m];
        v2f b1; b1.x = Wk0[16 + m]; b1.y = Wk0[64 + 16 + m];
        v2f b2; b2.x = Wk0[32 + m]; b2.y = Wk0[64 + 32 + m];
        v2f b3; b3.x = Wk0[48 + m]; b3.y = Wk0[64 + 48 + m];
        acc0 = __builtin_amdgcn_wmma_f32_16x16x4_f32(false, a, false, b0, (short)0, acc0, false, false);
        acc1 = __builtin_amdgcn_wmma_f32_16x16x4_f32(false, a, false, b1, (short)0, acc1, false, false);
        acc2 = __builtin_amdgcn_wmma_f32_16x16x4_f32(false, a, false, b2, (short)0, acc2, false, false);
        acc3 = __builtin_amdgcn_wmma_f32_16x16x4_f32(false, a, false, b3, (short)0, acc3, false, false);
    }
    #pragma unroll
    for (int r = 0; r < 8; ++r) {
        const size_t row = (size_t)(row0 + r + hi * 8);
        float* o = h + row * 64 + m;
        o[0]  = acc0[r];
        o[16] = acc1[r];
        o[32] = acc2[r];
        o[48] = acc3[r];
    }
}

// h2[100000,16] = relu(xe[100000,64]) @ W[64,16]
__global__ __launch_bounds__(128) void k_gemm2(const float* __restrict__ xe,
                                               const float* __restrict__ W,
                                               float* __restrict__ h2) {
    const int lane = threadIdx.x & 31;
    const int wave = threadIdx.x >> 5;
    const int tile = blockIdx.x * 4 + wave;
    if (tile >= NTILES) return;
    const int m  = lane & 15;
    const int hi = lane >> 4;
    const int row0 = tile * 16;

    v8f acc = {};
    const float* __restrict__ xr = xe + (size_t)(row0 + m) * 64;

    for (int k = 0; k < 64; k += 4) {
        const int ka = k + 2 * hi;
        v2f a; a.x = fmaxf(xr[ka], 0.0f); a.y = fmaxf(xr[ka + 1], 0.0f);
        v2f b; b.x = W[(size_t)ka * 16 + m]; b.y = W[(size_t)(ka + 1) * 16 + m];
        acc = __builtin_amdgcn_wmma_f32_16x16x4_f32(false, a, false, b, (short)0, acc, false, false);
    }
    #pragma unroll
    for (int r = 0; r < 8; ++r) {
        const size_t row = (size_t)(row0 + r + hi * 8);
        h2[row * 16 + m] = acc[r];
    }
}

// ---------------- self-loop + bias init, then edge scatter ----------------

// out[i*F+f] = dis[i]^2 * h[i*F+f] + b[f];  F = 1<<shiftF
__global__ __launch_bounds__(256) void k_selfinit(const float* __restrict__ dis,
                                                  const float* __restrict__ h,
                                                  const float* __restrict__ b,
                                                  float* __restrict__ out,
                                                  int shiftF, int total) {
    int idx = blockIdx.x * 256 + threadIdx.x;
    if (idx >= total) return;
    int i = idx >> shiftF;
    int f = idx & ((1 << shiftF) - 1);
    float d = dis[i];
    out[idx] = d * d * h[idx] + b[f];
}

// per-edge scatter-add: out[dst] += norm[e] * h[src]
// chunks/edge = 1<<shiftC (each chunk = float4), F = 1<<shiftF
__global__ __launch_bounds__(256) void k_scatter(const int* __restrict__ src,
                                                 const int* __restrict__ dst,
                                                 const float* __restrict__ norm,
                                                 const float* __restrict__ h,
                                                 float* __restrict__ out,
                                                 int shiftC, int shiftF, long long total) {
    long long t = (long long)blockIdx.x * 256 + threadIdx.x;
    if (t >= total) return;
    int e = (int)(t >> shiftC);
    int c = (int)(t & ((1 << shiftC) - 1));
    float nm = norm[e];
    int s = src[e], d = dst[e];
    const float4 v = *(const float4*)(h + ((size_t)s << shiftF) + c * 4);
    float* o = out + ((size_t)d << shiftF) + c * 4;
    atomicAdd(o + 0, nm * v.x);
    atomicAdd(o + 1, nm * v.y);
    atomicAdd(o + 2, nm * v.z);
    atomicAdd(o + 3, nm * v.w);
}

// ---------------- launch ----------------

extern "C" void kernel_launch(void* const* d_in, const int* in_sizes, int n_in,
                              void* d_out, int out_size, void* d_ws, size_t ws_size,
                              hipStream_t stream) {
    const float* x  = (const float*)d_in[0];               // [100000,128]
    const int*   ei = (const int*)d_in[1];                 // [2, 1600000]
    const float* ew = (const float*)d_in[2];               // [1600000]
    const float* W1 = (const float*)d_in[3];               // [128,64]
    const float* b1 = (const float*)d_in[4];               // [64]
    const float* W2 = (const float*)d_in[5];               // [64,16]
    const float* b2 = (const float*)d_in[6];               // [16]

    const int* src = ei;
    const int* dst = ei + N_EDGES;

    float* out   = (float*)d_out;                 // [100000,16]
    float* x_emb = (float*)d_out + N_NODES * 16;  // [100000,64]

    // workspace layout (floats)
    float* ws    = (float*)d_ws;
    float* dis   = ws;                       // 100000
    float* norm  = ws + 131072;              // 1600000
    float* h_lin = ws + 131072 + 1600000;    // up to 6400000 (reused for layer 2)

    const int nb_nodes = (N_NODES + 255) / 256;
    const int nb_edges = (N_EDGES + 255) / 256;
    const int nb_gemm  = (NTILES + 3) / 4;

    // normalization
    k_deg_init <<<nb_nodes, 256, 0, stream>>>(dis);
    k_deg_accum<<<nb_edges, 256, 0, stream>>>(dst, ew, dis);
    k_dis      <<<nb_nodes, 256, 0, stream>>>(dis);
    k_norm     <<<nb_edges, 256, 0, stream>>>(src, dst, ew, dis, norm);

    // layer 1: x @ W1 -> propagate -> x_emb (+b1)
    k_gemm1<<<nb_gemm, 128, 0, stream>>>(x, W1, h_lin);
    {
        int total = N_NODES * 64;
        k_selfinit<<<(total + 255) / 256, 256, 0, stream>>>(dis, h_lin, b1, x_emb, 6, total);
        long long tw = (long long)N_EDGES * 16;   // 16 float4-chunks per edge
        k_scatter<<<(int)((tw + 255) / 256), 256, 0, stream>>>(src, dst, norm, h_lin, x_emb, 4, 6, tw);
    }

    // layer 2: relu(x_emb) @ W2 -> propagate -> out (+b2)
    k_gemm2<<<nb_gemm, 128, 0, stream>>>(x_emb, W2, h_lin);   // h_lin reused as [100000,16]
    {
        int total = N_NODES * 16;
        k_selfinit<<<(total + 255) / 256, 256, 0, stream>>>(dis, h_lin, b2, out, 4, total);
        long long tw = (long long)N_EDGES * 4;    // 4 float4-chunks per edge
        k_scatter<<<(int)((tw + 255) / 256), 256, 0, stream>>>(src, dst, norm, h_lin, out, 2, 4, tw);
    }
}